// RelationalGraphConvolutionLP_40149354283031
// MI455X (gfx1250) — compile-verified
//
#include <hip/hip_runtime.h>

typedef __attribute__((ext_vector_type(2))) float v2f;
typedef __attribute__((ext_vector_type(4))) float v4f;
typedef __attribute__((ext_vector_type(8))) float v8f;

#define F 128
#define TILES_N 8          // 8 x 16 cols = 128 output cols per wave
#define WAVES_PER_BLOCK 8  // 256 threads, wave32
#define NREL_ORIG 5

// ---------------- utility kernels ----------------

__global__ void zero_u32(unsigned int* __restrict__ p, int n) {
    int i = blockIdx.x * blockDim.x + threadIdx.x;
    if (i < n) p[i] = 0u;
}

// Per-(relation,node) in-degree counts for both directions, plus per-relation
// bucket sizes (fused so the edge list is read once during setup).
__global__ void count_edges(const int* __restrict__ graph, float* __restrict__ c,
                            int* __restrict__ relCnt, int E, int N) {
    int e = blockIdx.x * blockDim.x + threadIdx.x;
    if (e >= E) return;
    int s = graph[3 * e + 0];
    int r = graph[3 * e + 1];
    int d = graph[3 * e + 2];
    atomicAdd(&c[(size_t)r * N + d], 1.0f);
    atomicAdd(&c[(size_t)(r + NREL_ORIG) * N + s], 1.0f);
    atomicAdd(&relCnt[r], 1);
}

__global__ void recip_counts(float* __restrict__ c, int n) {
    int i = blockIdx.x * blockDim.x + threadIdx.x;
    if (i < n) c[i] = 1.0f / fmaxf(c[i], 1.0f);
}

// Tiny serial prefix over NREL_ORIG bucket counts (device-side: capture-safe).
__global__ void prefix_rel(const int* __restrict__ relCnt,
                           int* __restrict__ relStart, int* __restrict__ cursor) {
    if (threadIdx.x == 0 && blockIdx.x == 0) {
        int acc = 0;
        for (int r = 0; r < NREL_ORIG; ++r) {
            relStart[r] = acc;
            cursor[r] = acc;
            acc += relCnt[r];
        }
        relStart[NREL_ORIG] = acc;
    }
}

// Counting-sort placement: bucket[] holds edge indices grouped by relation.
__global__ void bucket_place(const int* __restrict__ graph, int* __restrict__ cursor,
                             int* __restrict__ bucket, int E) {
    int e = blockIdx.x * blockDim.x + threadIdx.x;
    if (e >= E) return;
    int r = graph[3 * e + 1];
    int pos = atomicAdd(&cursor[r], 1);
    bucket[pos] = e;
}

// ---------------- dense GEMM: Out(MxF) = X(MxF) @ (B0 + B1)(FxF) ----------------
// Wave32 WMMA f32 16x16x4. Each wave computes a 16-row x 128-col strip.
// LDS holds the weight in a K-quad swizzled layout so each WMMA B operand pair
// (B[kl][n], B[kl+1][n]) is one aligned, bank-conflict-free ds_load_b64:
//   Bs[(k>>2)*512 + n*4 + (k&3)] = B[k][n]
__global__ __launch_bounds__(256) void gemm_x_w(
    const float* __restrict__ X, const float* __restrict__ B0,
    const float* __restrict__ B1, float* __restrict__ Out, int M) {

    __shared__ float Bs[F * F];  // 64 KB

    // Cooperative staging with swizzle (optionally fused add of second matrix).
    if (B1 != nullptr) {
        const v4f* b0v = (const v4f*)B0;
        const v4f* b1v = (const v4f*)B1;
        for (int j = threadIdx.x; j < (F * F) / 4; j += 256) {
            v4f val = b0v[j] + b1v[j];
            int k = (j * 4) / F;
            int n = (j * 4) % F;
            float* dst = &Bs[((k >> 2) << 9) + (k & 3)];
            dst[(n + 0) * 4] = val.x;
            dst[(n + 1) * 4] = val.y;
            dst[(n + 2) * 4] = val.z;
            dst[(n + 3) * 4] = val.w;
        }
    } else {
        const v4f* b0v = (const v4f*)B0;
        for (int j = threadIdx.x; j < (F * F) / 4; j += 256) {
            v4f val = b0v[j];
            int k = (j * 4) / F;
            int n = (j * 4) % F;
            float* dst = &Bs[((k >> 2) << 9) + (k & 3)];
            dst[(n + 0) * 4] = val.x;
            dst[(n + 1) * 4] = val.y;
            dst[(n + 2) * 4] = val.z;
            dst[(n + 3) * 4] = val.w;
        }
    }
    __syncthreads();

    const int wave = threadIdx.x >> 5;
    const int lane = threadIdx.x & 31;
    const int tile = blockIdx.x * WAVES_PER_BLOCK + wave;  // 16-row tile index
    const int mtiles = M >> 4;
    if (tile >= mtiles) return;  // wave-uniform: EXEC stays all-ones for WMMA

    const int col   = lane & 15;          // N-index within tile / A-matrix row slot
    const int khalf = (lane >> 4) << 1;   // lanes 16..31 handle K = k0+2, k0+3
    const int row   = (tile << 4) + col;

    v8f acc[TILES_N];
#pragma unroll
    for (int t = 0; t < TILES_N; ++t)
#pragma unroll
        for (int v = 0; v < 8; ++v) acc[t][v] = 0.0f;

    const float* xrow = X + (size_t)row * F;

    for (int k0 = 0; k0 < F; k0 += 4) {
        const int kl = k0 + khalf;
        // A 16x4 layout: VGPR0 = X[row, kl], VGPR1 = X[row, kl+1]  (8B aligned)
        const v2f a = *(const v2f*)(xrow + kl);
        // Swizzled B base for this K-quad and lane-half.
        const float* brow = &Bs[((k0 >> 2) << 9) + khalf];
#pragma unroll
        for (int t = 0; t < TILES_N; ++t) {
            // Single ds_load_b64: (B[kl][n], B[kl+1][n]) contiguous by layout.
            const v2f b = *(const v2f*)(brow + (t * 16 + col) * 4);
            acc[t] = __builtin_amdgcn_wmma_f32_16x16x4_f32(
                false, a, false, b, (short)0, acc[t], false, false);
        }
    }

    // C/D layout: VGPR v -> lanes 0-15: M=v, lanes 16-31: M=v+8; N = lane&15
    const int rbase = (tile << 4) + ((lane >> 4) << 3);
#pragma unroll
    for (int t = 0; t < TILES_N; ++t) {
#pragma unroll
        for (int v = 0; v < 8; ++v) {
            Out[(size_t)(rbase + v) * F + t * 16 + col] = acc[t][v];
        }
    }
}

// ---------------- scatter: out[dst] += Z[src] * invc[relIdx, dst] ----------------
// Grid-stride over one relation's bucket; one wave per directed edge iteration,
// lane l owns 4 consecutive floats (single b128 gather + 4 f32 atomics).
__global__ __launch_bounds__(256) void scatter_edges(
    const int* __restrict__ graph, const int* __restrict__ bucket,
    const int* __restrict__ relStart, const float* __restrict__ Z,
    const float* __restrict__ invc, float* __restrict__ out,
    int N, int rsel, int rev, int relIdx) {

    const int base = relStart[rsel];
    const int nb   = relStart[rsel + 1] - base;

    const int lane = threadIdx.x & 31;
    const int wid  = (int)((blockIdx.x * blockDim.x + threadIdx.x) >> 5);
    const int nw   = (int)((gridDim.x * blockDim.x) >> 5);

    for (int i = wid; i < nb; i += nw) {
        const int e = bucket[base + i];
        int s = graph[3 * e + 0];
        int d = graph[3 * e + 2];
        if (rev) { int t = s; s = d; d = t; }

        const float inv = invc[(size_t)relIdx * N + d];
        const v4f z = *(const v4f*)(Z + (size_t)s * F + lane * 4);
        float* od = out + (size_t)d * F + lane * 4;
        atomicAdd(od + 0, z.x * inv);
        atomicAdd(od + 1, z.y * inv);
        atomicAdd(od + 2, z.z * inv);
        atomicAdd(od + 3, z.w * inv);
    }
}

// ---------------- launch ----------------

extern "C" void kernel_launch(void* const* d_in, const int* in_sizes, int n_in,
                              void* d_out, int out_size, void* d_ws, size_t ws_size,
                              hipStream_t stream) {
    const int*   graph = (const int*)d_in[0];    // (E,3) int32
    const float* X     = (const float*)d_in[1];  // (N,128) f32
    const float* W     = (const float*)d_in[2];  // (11,128,128) f32
    const float* root  = (const float*)d_in[3];  // (128,128) f32
    float* out = (float*)d_out;

    const int E = in_sizes[0] / 3;
    const int N = in_sizes[1] / F;
    const int cn = 2 * NREL_ORIG * N;  // 10 * N reciprocal-count slots

    // Workspace layout (256B-aligned chunks):
    char* wp = (char*)d_ws;
    auto take = [&](size_t bytes) {
        char* p = wp;
        wp += (bytes + 255) & ~(size_t)255;
        return p;
    };
    float* c        = (float*)take((size_t)cn * 4);
    int*   relCnt   = (int*)take(NREL_ORIG * 4);
    int*   relStart = (int*)take((NREL_ORIG + 1) * 4);
    int*   cursor   = (int*)take(NREL_ORIG * 4);
    int*   bucket   = (int*)take((size_t)E * 4);
    float* Z        = (float*)take((size_t)N * F * 4);

    // 1) zero counts
    zero_u32<<<(cn + 255) / 256, 256, 0, stream>>>((unsigned int*)c, cn);
    zero_u32<<<1, 64, 0, stream>>>((unsigned int*)relCnt, NREL_ORIG);

    // 2) per-(rel,node) degree + per-rel bucket sizes
    count_edges<<<(E + 255) / 256, 256, 0, stream>>>(graph, c, relCnt, E, N);
    recip_counts<<<(cn + 255) / 256, 256, 0, stream>>>(c, cn);

    // 3) bucket edges by relation (counting sort, device-side offsets)
    prefix_rel<<<1, 1, 0, stream>>>(relCnt, relStart, cursor);
    bucket_place<<<(E + 255) / 256, 256, 0, stream>>>(graph, cursor, bucket, E);

    const int mtiles = N / 16;
    const int gblocks = (mtiles + WAVES_PER_BLOCK - 1) / WAVES_PER_BLOCK;

    // 4) self-loop relation has count==1 per node -> out = X @ (W[10] + root)
    gemm_x_w<<<gblocks, 256, 0, stream>>>(
        X, W + (size_t)2 * NREL_ORIG * F * F, root, out, N);

    // 5) 10 real relations: dense WMMA transform then mean-weighted scatter
    const int sblocks = 256;  // 2048 grid-stride waves per pass
    for (int p = 0; p < 2 * NREL_ORIG; ++p) {
        const float* Wr = W + (size_t)p * F * F;
        gemm_x_w<<<gblocks, 256, 0, stream>>>(X, Wr, nullptr, Z, N);
        scatter_edges<<<sblocks, 256, 0, stream>>>(
            graph, bucket, relStart, Z, c, out,
            N, p % NREL_ORIG, (p >= NREL_ORIG) ? 1 : 0, p);
    }
}